// LIANNmodel_29935922053500
// MI455X (gfx1250) — compile-verified
//
#include <hip/hip_runtime.h>

// MI455X / gfx1250, wave32. f32 WMMA (V_WMMA_F32_16X16X4_F32) for exact-fp32
// matrix math; max/argmax-over-S fused into the GEMM tile loop; Hebbian term
// rewritten as histogram-weighted W0 sweep (O(H*S*D) instead of O(B*H*D)).

typedef float v2f __attribute__((ext_vector_type(2)));
typedef float v8f __attribute__((ext_vector_type(8)));

#define B_ 512
#define S_ 32
#define H_ 512
#define D_ 256
#define C_ 128
#define LIANN_LR 0.001f

// ---------------------------------------------------------------------------
// Kernel 1: fused  z[b,s,h] = x·W0^T + b0  ->  top-1 over s (max + argmax)
// grid = (B/16, H/128), block = 256 threads (8 waves). Each wave: one 16x16
// (m-rows x h-cols) tile, loops s=0..31, K-loop of 64 f32-WMMAs each.
// Outputs: hAct[b,h] = relu(max_s z), cnt[h,s] += votes(argmax==s).
// ---------------------------------------------------------------------------
__global__ __launch_bounds__(256) void k_gemm0_topk(
    const float* __restrict__ x, const float* __restrict__ W0,
    const float* __restrict__ b0, float* __restrict__ hAct,
    int* __restrict__ cnt)
{
  __shared__ float xs[16 * 258];            // padded stride vs bank conflicts
  const int m0  = blockIdx.x * 16;
  const int tid = threadIdx.x;

  // stage x tile [16][256] into LDS once; reused across all 32 s-iterations
  for (int i = tid; i < 16 * 256; i += 256) {
    int r = i >> 8, c = i & 255;
    xs[r * 258 + c] = x[(m0 + r) * D_ + c];
  }
  __syncthreads();

  const int wave = tid >> 5;
  const int lane = tid & 31;
  const int ln   = lane & 15;               // N (and A-row M) sub-index
  const int hi2  = (lane >> 4) * 2;         // K sub-offset: lanes16-31 -> K+2
  const int h0   = (blockIdx.y * 8 + wave) * 16;

  v8f zmax;
  int amax[8];

  for (int s = 0; s < S_; ++s) {
    v8f acc = {};
    const float* w0row = W0 + (size_t)(s * H_ + h0 + ln) * D_ + hi2;  // B-frag
    const float* xrow  = xs + ln * 258 + hi2;                          // A-frag
    #pragma unroll 8
    for (int k0 = 0; k0 < D_; k0 += 4) {
      v2f a = *(const v2f*)(xrow + k0);      // A[m=ln][k0+hi2 .. +1]  (LDS)
      v2f b = *(const v2f*)(w0row + k0);     // B[k][n=ln] = W0[n][k]  (global)
      acc = __builtin_amdgcn_wmma_f32_16x16x4_f32(
          false, a, false, b, (short)0, acc, false, false);
    }
    float bias = b0[s * H_ + h0 + ln];       // bias depends on column only
    if (s == 0) {
      #pragma unroll
      for (int v = 0; v < 8; ++v) { zmax[v] = acc[v] + bias; amax[v] = 0; }
    } else {
      #pragma unroll
      for (int v = 0; v < 8; ++v) {
        float zv = acc[v] + bias;
        if (zv > zmax[v]) { zmax[v] = zv; amax[v] = s; }  // strict > keeps first
      }
    }
  }

  // C/D layout: VGPR v -> row m0 + v + 8*(lane>=16), col h0 + (lane&15)
  const int hcol  = h0 + ln;
  const int rbase = m0 + ((lane >> 4) << 3);
  #pragma unroll
  for (int v = 0; v < 8; ++v) {
    float hv = zmax[v] > 0.f ? zmax[v] : 0.f;        // relu(xTop)
    hAct[(rbase + v) * H_ + hcol] = hv;
    atomicAdd(&cnt[hcol * S_ + amax[v]], 1);          // argmax histogram
  }
}

// ---------------------------------------------------------------------------
// Kernel 2: meanUpd[d] = LR/B * sum_b sign(x[b,d])   (active - inactive)
// ---------------------------------------------------------------------------
__global__ void k_meanupd(const float* __restrict__ x, float* __restrict__ mu)
{
  int d = blockIdx.x * blockDim.x + threadIdx.x;
  if (d >= D_) return;
  float s = 0.f;
  for (int b = 0; b < B_; ++b) {
    float v = x[b * D_ + d];
    s += (v > 0.f) ? 1.f : ((v < 0.f) ? -1.f : 0.f);
  }
  mu[d] = s * (LIANN_LR / (float)B_);
}

// ---------------------------------------------------------------------------
// Kernel 3: topKweights[h,d] = (1/B) * sum_s cnt[h,s]*W0[s,h,d] + meanUpd[d]
// grid = H blocks, 256 threads = one d each (D==256). W0 swept once (16 MB).
// ---------------------------------------------------------------------------
__global__ __launch_bounds__(256) void k_topkw(
    const float* __restrict__ W0, const int* __restrict__ cnt,
    const float* __restrict__ mu, float* __restrict__ outw)
{
  const int h = blockIdx.x;
  const int d = threadIdx.x;
  __shared__ float c_sh[S_];
  if (d < S_) c_sh[d] = (float)cnt[h * S_ + d];
  __syncthreads();
  float acc = 0.f;
  #pragma unroll 8
  for (int s = 0; s < S_; ++s)
    acc += c_sh[s] * W0[(size_t)(s * H_ + h) * D_ + d];
  outw[h * D_ + d] = acc * (1.f / (float)B_) + mu[d];
}

// ---------------------------------------------------------------------------
// Kernel 4: logits = relu(xTop)·W1^T + b1   (M=512, N=128, K=512), f32 WMMA.
// grid = B/16 blocks, 8 waves cover all 8 c-tiles of C=128.
// ---------------------------------------------------------------------------
__global__ __launch_bounds__(256) void k_gemm1(
    const float* __restrict__ hAct, const float* __restrict__ W1,
    const float* __restrict__ b1, float* __restrict__ logits)
{
  __shared__ float hs[16 * 514];
  const int m0  = blockIdx.x * 16;
  const int tid = threadIdx.x;
  for (int i = tid; i < 16 * 512; i += 256) {
    int r = i >> 9, c = i & 511;
    hs[r * 514 + c] = hAct[(m0 + r) * H_ + c];
  }
  __syncthreads();

  const int wave = tid >> 5, lane = tid & 31;
  const int ln = lane & 15, hi2 = (lane >> 4) * 2;
  const int c0 = wave * 16;

  v8f acc = {};
  const float* wrow = W1 + (c0 + ln) * H_ + hi2;
  const float* arow = hs + ln * 514 + hi2;
  #pragma unroll 8
  for (int k0 = 0; k0 < H_; k0 += 4) {
    v2f a = *(const v2f*)(arow + k0);
    v2f b = *(const v2f*)(wrow + k0);
    acc = __builtin_amdgcn_wmma_f32_16x16x4_f32(
        false, a, false, b, (short)0, acc, false, false);
  }
  float bias = b1[c0 + ln];
  const int rbase = m0 + ((lane >> 4) << 3);
  #pragma unroll
  for (int v = 0; v < 8; ++v)
    logits[(rbase + v) * C_ + c0 + ln] = acc[v] + bias;
}

// ---------------------------------------------------------------------------
// Kernel 5: per-row log_softmax + NLL loss + accuracy (atomics into d_out[0,1])
// grid = B blocks, C=128 threads.
// ---------------------------------------------------------------------------
__global__ __launch_bounds__(128) void k_softmax_loss(
    const float* __restrict__ logits, const int* __restrict__ y,
    float* __restrict__ out)
{
  const int b = blockIdx.x, c = threadIdx.x;
  __shared__ float rv[C_];
  __shared__ int   ri[C_];
  float l = logits[b * C_ + c];

  // (max, first-argmax) reduction
  rv[c] = l; ri[c] = c;
  __syncthreads();
  for (int off = C_ / 2; off > 0; off >>= 1) {
    if (c < off) {
      float v2 = rv[c + off]; int i2 = ri[c + off];
      if (v2 > rv[c] || (v2 == rv[c] && i2 < ri[c])) { rv[c] = v2; ri[c] = i2; }
    }
    __syncthreads();
  }
  float mx = rv[0]; int am = ri[0];
  __syncthreads();

  // sum(exp) reduction
  rv[c] = expf(l - mx);
  __syncthreads();
  for (int off = C_ / 2; off > 0; off >>= 1) {
    if (c < off) rv[c] += rv[c + off];
    __syncthreads();
  }
  float lp = l - mx - logf(rv[0]);

  out[2 + b * C_ + c] = lp;                       // logp block of d_out
  int t = y[b];
  if (c == t)            atomicAdd(&out[0], -lp / (float)B_);   // loss
  if (c == 0 && am == t) atomicAdd(&out[1], 1.f / (float)B_);   // accuracy
}

// ---------------------------------------------------------------------------
// Kernel 0: zero the atomic accumulators (loss, acc) and the vote histogram.
// ---------------------------------------------------------------------------
__global__ void k_init(float* __restrict__ out, int* __restrict__ cnt)
{
  int i = blockIdx.x * blockDim.x + threadIdx.x;
  if (i < 2) out[i] = 0.f;
  if (i < H_ * S_) cnt[i] = 0;
}

extern "C" void kernel_launch(void* const* d_in, const int* in_sizes, int n_in,
                              void* d_out, int out_size, void* d_ws, size_t ws_size,
                              hipStream_t stream)
{
  (void)in_sizes; (void)n_in; (void)out_size; (void)ws_size;
  const float* x  = (const float*)d_in[0];
  const int*   y  = (const int*)  d_in[1];
  const float* W0 = (const float*)d_in[2];
  const float* b0 = (const float*)d_in[3];
  const float* W1 = (const float*)d_in[4];
  const float* b1 = (const float*)d_in[5];
  float* out = (float*)d_out;

  // workspace carve (~1.4 MB): hAct | cnt | meanUpd | logits
  float* hAct    = (float*)d_ws;                       // B*H
  int*   cnt     = (int*)(hAct + (size_t)B_ * H_);     // H*S
  float* meanUpd = (float*)(cnt + H_ * S_);            // D
  float* logits  = meanUpd + D_;                       // B*C

  float* out_topkw = out + 2 + (size_t)B_ * C_;        // [H,D] block of d_out

  k_init<<<(H_ * S_ + 255) / 256, 256, 0, stream>>>(out, cnt);
  k_gemm0_topk<<<dim3(B_ / 16, H_ / 128), 256, 0, stream>>>(x, W0, b0, hAct, cnt);
  k_meanupd<<<1, 256, 0, stream>>>(x, meanUpd);
  k_topkw<<<H_, 256, 0, stream>>>(W0, cnt, meanUpd, out_topkw);
  k_gemm1<<<B_ / 16, 256, 0, stream>>>(hAct, W1, b1, logits);
  k_softmax_loss<<<B_, C_, 0, stream>>>(logits, y, out);
}